// SmartPostProcess_3908420239603
// MI455X (gfx1250) — compile-verified
//
#include <hip/hip_runtime.h>

typedef __attribute__((ext_vector_type(2))) float v2f;
typedef __attribute__((ext_vector_type(8))) float v8f;

#define VOL_D 192
#define VOL_H 256
#define VOL_W 256
#define VOL_N (VOL_D * VOL_H * VOL_W)   // 12,582,912

// Gaussian kernels, precomputed: r = int(3*sigma + 0.5)
// sigma=1 -> r=3 (7 taps), sigma=2 -> r=6 (13 taps)
__constant__ float GW_S1[7] = {
    0.004433048f, 0.054005582f, 0.242036411f, 0.399050319f,
    0.242036411f, 0.054005582f, 0.004433048f};
__constant__ float GW_S2[13] = {
    0.002218195f, 0.008773137f, 0.027023170f, 0.064825175f,
    0.121109427f, 0.176213134f, 0.199675571f, 0.176213134f,
    0.121109427f, 0.064825175f, 0.027023170f, 0.008773137f,
    0.002218195f};

__device__ __forceinline__ int iclamp(int v, int lo, int hi) {
  return min(max(v, lo), hi);
}

// ---------------------------------------------------------------------------
// 1-D Gaussian blur along an arbitrary axis as WMMA GEMMs:
//   out(16x16) = Band(16xK) x Slab(Kx16),  Band[m][k] = w[k-m]
// One wave per 16x16 tile. All strides/radius are template constants so the
// chunk loop fully unrolls and index math is 32-bit. All edge handling is
// branchless: unconditional clamped load, then v_cndmask select of zero.
// The A (band) operands are tile-invariant and hoisted before the main loop,
// so the inner loop is pure {2 B-loads -> select -> v_wmma} per K-chunk.
// Layouts per CDNA5 ISA 7.12.2 (f32 16x16x4):
//   A: lane<16 -> (M=lane, K=kc+0/+1), lane>=16 -> (M=lane-16, K=kc+2/+3)
//   B: lane<16 -> (K=kc+0/+1, N=lane), lane>=16 -> (K=kc+2/+3, N=lane-16)
//   C: lane<16: N=lane, VGPR v -> M=v ; lane>=16: N=lane-16, M=v+8
// ---------------------------------------------------------------------------
template <int SM, int SN, int SO, int LENA, int R>
__global__ __launch_bounds__(256) void blur_axis_wmma(
    const float* __restrict__ in, float* __restrict__ out,
    int nMt, int nNt, int nTiles)
{
  const int wave = threadIdx.x >> 5;
  const int lane = threadIdx.x & 31;
  const int tile = blockIdx.x * (blockDim.x >> 5) + wave;
  if (tile >= nTiles) return;                 // wave-uniform: EXEC stays full

  const float* wt = (R == 3) ? GW_S1 : GW_S2;
  constexpr int NCHUNK = (16 + 2 * R + 3) >> 2;   // 6 (R=3) or 7 (R=6)

  const int laneN  = lane & 15;
  const int hiHalf = lane >> 4;               // 0 or 1
  const int mA     = lane & 15;               // A-matrix row for this lane

  // ---- Tile-invariant A (band) operands: unconditional loads + select ----
  v2f aCh[NCHUNK];
#pragma unroll
  for (int ch = 0; ch < NCHUNK; ++ch) {
    const int kc  = ch * 4 + hiHalf * 2;
    const int d0  = kc - mA,            d1  = kc + 1 - mA;
    const int d0c = iclamp(d0, 0, 2*R), d1c = iclamp(d1, 0, 2*R);
    const float w0 = wt[d0c];
    const float w1 = wt[d1c];
    aCh[ch].x = (d0 == d0c) ? w0 : 0.0f;
    aCh[ch].y = (d1 == d1c) ? w1 : 0.0f;
  }

  const int tm = tile % nMt;
  const int tq = tile / nMt;
  const int tn = tq % nNt;
  const int o  = tq / nNt;

  const int mStart = tm * 16;
  const int baseN  = o * SO + (tn * 16 + laneN) * SN;

  v8f c = {};
#pragma unroll
  for (int ch = 0; ch < NCHUNK; ++ch) {
    const int kc  = ch * 4 + hiHalf * 2;      // first K this lane holds
    const int ca0 = mStart - R + kc;
    const int ca1 = ca0 + 1;
    const int cc0 = iclamp(ca0, 0, LENA - 1);
    const int cc1 = iclamp(ca1, 0, LENA - 1);
    const float b0 = in[baseN + cc0 * SM];
    const float b1 = in[baseN + cc1 * SM];
    v2f b;
    b.x = (ca0 == cc0) ? b0 : 0.0f;
    b.y = (ca1 == cc1) ? b1 : 0.0f;
    c = __builtin_amdgcn_wmma_f32_16x16x4_f32(
        /*neg_a=*/false, aCh[ch], /*neg_b=*/false, b,
        /*c_mod=*/(short)0, c, /*reuse_a=*/false, /*reuse_b=*/false);
  }

  const int mOff = hiHalf * 8;
  if constexpr (SM == 1) {
    // Contiguous along M: two 128-bit stores (32B-aligned: mStart+mOff % 8 == 0)
    float4 lo = make_float4(c[0], c[1], c[2], c[3]);
    float4 hi = make_float4(c[4], c[5], c[6], c[7]);
    float4* p = (float4*)(out + baseN + mStart + mOff);
    p[0] = lo;
    p[1] = hi;
  } else {
#pragma unroll
    for (int v = 0; v < 8; ++v)
      out[baseN + (mStart + mOff + v) * SM] = c[v];
  }
}

// ---------------------------------------------------------------------------
// Hessian + approximate eigenvalues + surfaceness (pointwise, 3D halo).
// Branchless zero-padded fetch: clamped unconditional load + cndmask.
// ---------------------------------------------------------------------------
__device__ __forceinline__ float vat(const float* __restrict__ t,
                                     int d, int h, int w) {
  const int dc = iclamp(d, 0, VOL_D - 1);
  const int hc = iclamp(h, 0, VOL_H - 1);
  const int wc = iclamp(w, 0, VOL_W - 1);
  const float v = t[(dc << 16) + (hc << 8) + wc];
  return (d == dc && h == hc && w == wc) ? v : 0.0f;
}

__global__ __launch_bounds__(256) void hess_surf(
    const float* __restrict__ t, float* __restrict__ surf, float s2, int first)
{
  const int i = blockIdx.x * blockDim.x + threadIdx.x;
  if (i >= VOL_N) return;
  const int w = i & 255, h = (i >> 8) & 255, d = i >> 16;

  const float f0 = t[i];
  const float fxx = (vat(t,d+1,h,w) + vat(t,d-1,h,w) - 2.0f*f0) * s2;
  const float fyy = (vat(t,d,h+1,w) + vat(t,d,h-1,w) - 2.0f*f0) * s2;
  const float fzz = (vat(t,d,h,w+1) + vat(t,d,h,w-1) - 2.0f*f0) * s2;
  const float fxy = 0.25f*(vat(t,d+1,h+1,w)-vat(t,d+1,h-1,w)
                          -vat(t,d-1,h+1,w)+vat(t,d-1,h-1,w))*s2;
  const float fxz = 0.25f*(vat(t,d+1,h,w+1)-vat(t,d+1,h,w-1)
                          -vat(t,d-1,h,w+1)+vat(t,d-1,h,w-1))*s2;
  const float fyz = 0.25f*(vat(t,d,h+1,w+1)-vat(t,d,h+1,w-1)
                          -vat(t,d,h-1,w+1)+vat(t,d,h-1,w-1))*s2;

  float v1 = fxx + fxy + fxz;
  float v2 = fxy + fyy + fyz;
  float v3 = fxz + fyz + fzz;
  const float vn = sqrtf(v1*v1 + v2*v2 + v3*v3 + 1e-7f);
  v1 /= vn; v2 /= vn; v3 /= vn;
  const float Hv1 = fxx*v1 + fxy*v2 + fxz*v3;
  const float Hv2 = fxy*v1 + fyy*v2 + fyz*v3;
  const float Hv3 = fxz*v1 + fyz*v2 + fzz*v3;
  const float l3  = Hv1*v1 + Hv2*v2 + Hv3*v3;
  const float trace = fxx + fyy + fzz;
  const float det = fxx*(fyy*fzz - fyz*fyz)
                  - fxy*(fxy*fzz - fxz*fyz)
                  + fxz*(fxy*fyz - fxz*fyy);
  const float s  = trace - l3;
  const float p  = det / (l3 + 1e-7f);
  const float disc = fmaxf(s*s - 4.0f*p, 0.0f);
  const float sd = sqrtf(disc + 1e-7f);
  float e0 = 0.5f*(s - sd), e1 = 0.5f*(s + sd), e2 = l3;
  // sort ascending by |e| (3-element network)
  if (fabsf(e1) < fabsf(e0)) { float tt = e0; e0 = e1; e1 = tt; }
  if (fabsf(e2) < fabsf(e1)) { float tt = e1; e1 = e2; e2 = tt; }
  if (fabsf(e1) < fabsf(e0)) { float tt = e0; e0 = e1; e1 = tt; }

  const float plate = expf(-(e0*e0 + e1*e1) / (0.25f*(e2*e2 + 1e-7f)));
  const float sv = plate * fabsf(e2);
  surf[i] = first ? sv : fmaxf(surf[i], sv);
}

// ---------------------------------------------------------------------------
// Global max (surf >= 0 so uint-bit atomicMax is order-preserving).
// ---------------------------------------------------------------------------
__global__ void init_max(unsigned* slot) { *slot = 0u; }

__global__ __launch_bounds__(256) void reduce_max(
    const float* __restrict__ surf, unsigned* slot)
{
  __shared__ float sm[256];
  float m = 0.0f;
  for (int i = blockIdx.x * blockDim.x + threadIdx.x; i < VOL_N;
       i += gridDim.x * blockDim.x)
    m = fmaxf(m, surf[i]);
  sm[threadIdx.x] = m;
  __syncthreads();
  for (int s = 128; s > 0; s >>= 1) {
    if (threadIdx.x < s)
      sm[threadIdx.x] = fmaxf(sm[threadIdx.x], sm[threadIdx.x + s]);
    __syncthreads();
  }
  if (threadIdx.x == 0) atomicMax(slot, __float_as_uint(sm[0]));
}

__global__ __launch_bounds__(256) void finalize(
    const float* __restrict__ in, float* __restrict__ out,
    const unsigned* __restrict__ slot)
{
  const int i = blockIdx.x * blockDim.x + threadIdx.x;
  if (i >= VOL_N) return;
  const float M  = __uint_as_float(*slot);
  const float sf = out[i] / (M + 1e-7f);
  const float v  = in[i] + 0.3f * sf;
  float r = 1.0f / (1.0f + expf(-8.0f * (v - 0.5f)));
  out[i] = fminf(fmaxf(r, 0.0f), 1.0f);
}

// ---------------------------------------------------------------------------
// Pipeline: per sigma { blurD, blurH, blurW (WMMA) -> hess+surf } -> max -> final
// Scratch: two ping-pong volumes (2 x 48 MiB) + 1 dword; surf lives in d_out.
// input + t0 + t1 + d_out ~ 192 MiB -> intermediate sweeps stay in MI455X L2.
// ---------------------------------------------------------------------------
extern "C" void kernel_launch(void* const* d_in, const int* in_sizes, int n_in,
                              void* d_out, int out_size, void* d_ws, size_t ws_size,
                              hipStream_t stream) {
  (void)in_sizes; (void)n_in; (void)out_size; (void)ws_size;
  const float* vol  = (const float*)d_in[0];
  float*       surf = (float*)d_out;
  float*       t0   = (float*)d_ws;
  float*       t1   = t0 + VOL_N;
  unsigned*    slot = (unsigned*)(t1 + VOL_N);

  init_max<<<1, 1, 0, stream>>>(slot);

  const dim3 blk(256);
  const int nTiles = VOL_N / 256;                 // 49152 tiles (16x16 each)
  const int tgrid  = (nTiles + 7) / 8;            // 8 waves per block
  const int pgrid  = (VOL_N + 255) / 256;

  // sigma = 1 (r = 3)
  blur_axis_wmma<65536, 1, 256, VOL_D, 3><<<tgrid, blk, 0, stream>>>(
      vol, t0, 12, 16, nTiles);                   // D axis
  blur_axis_wmma<256, 1, 65536, VOL_H, 3><<<tgrid, blk, 0, stream>>>(
      t0, t1, 16, 16, nTiles);                    // H axis
  blur_axis_wmma<1, 256, 65536, VOL_W, 3><<<tgrid, blk, 0, stream>>>(
      t1, t0, 16, 16, nTiles);                    // W axis
  hess_surf<<<pgrid, blk, 0, stream>>>(t0, surf, 1.0f, 1);

  // sigma = 2 (r = 6)
  blur_axis_wmma<65536, 1, 256, VOL_D, 6><<<tgrid, blk, 0, stream>>>(
      vol, t0, 12, 16, nTiles);                   // D axis
  blur_axis_wmma<256, 1, 65536, VOL_H, 6><<<tgrid, blk, 0, stream>>>(
      t0, t1, 16, 16, nTiles);                    // H axis
  blur_axis_wmma<1, 256, 65536, VOL_W, 6><<<tgrid, blk, 0, stream>>>(
      t1, t0, 16, 16, nTiles);                    // W axis
  hess_surf<<<pgrid, blk, 0, stream>>>(t0, surf, 4.0f, 0);

  reduce_max<<<1024, blk, 0, stream>>>(surf, slot);
  finalize<<<pgrid, blk, 0, stream>>>(vol, surf, slot);
}